// PatchMasker_26577257627890
// MI455X (gfx1250) — compile-verified
//
#include <hip/hip_runtime.h>
#include <hip/hip_bf16.h>

// ---------------------------------------------------------------------------
// PatchMasker for MI455X (gfx1250).
// Pure bandwidth problem: ~205 MiB of traffic (skip reads of masked rows),
// ~9us at 23.3 TB/s. Uses CDNA5 Tensor Data Mover (tensor_load_to_lds) to DMA
// unmasked rows into LDS, s_wait_tensorcnt to sync, and non-temporal B128
// stores for the 128 MiB output stream.
// ---------------------------------------------------------------------------

typedef unsigned int u32x4 __attribute__((ext_vector_type(4)));
typedef int          i32x8 __attribute__((ext_vector_type(8)));
typedef int          i32x4 __attribute__((ext_vector_type(4)));
typedef float        f32x4 __attribute__((ext_vector_type(4)));

#define N_PATCH   512
#define N_MASK    204          // int(0.4 * 512)
#define FEAT      1024
#define N_ROWS    (64 * 512)   // batch * patches = 32768 rows of 4KB
#define WAVES_PER_WG   4
#define ROWS_PER_WAVE  4
#define ROWS_PER_WG    (WAVES_PER_WG * ROWS_PER_WAVE)   // 16

__device__ __forceinline__ unsigned rotl32(unsigned x, int r) {
    return (x << r) | (x >> (32 - r));
}

__device__ __forceinline__ unsigned rfl(unsigned v) {
    return (unsigned)__builtin_amdgcn_readfirstlane((int)v);
}

// --------------------------------------------------------------------------
// Apply kernel (defined first so its asm leads the disassembly snippet).
// Each wave owns ROWS_PER_WAVE rows; unmasked rows are DMA'd global->LDS with
// the Tensor Data Mover, then written out with non-temporal B128 stores.
// Masked rows: store zeros immediately (before the tensorcnt wait), never
// read x.
// --------------------------------------------------------------------------
__global__ void __launch_bounds__(WAVES_PER_WG * 32)
PatchMasker_apply_kernel(const float* __restrict__ x,
                         const unsigned* __restrict__ mask,
                         float* __restrict__ out) {
    __shared__ float tile[WAVES_PER_WG][ROWS_PER_WAVE][FEAT];   // 64 KB

    const int      lane = threadIdx.x & 31;
    const unsigned wave = rfl((unsigned)threadIdx.x >> 5);      // SGPR-uniform
    const unsigned baseRow = (unsigned)blockIdx.x * ROWS_PER_WG + wave * ROWS_PER_WAVE;

    // D# group1 (constant for every row): workgroup_mask=0, data_size=2 (4B),
    // tensor_dim0=1024, tensor_dim1=1, tile_dim0=1024, tile_dim1=1,
    // tensor_dim0_stride=1024, everything else 0.
    const i32x8 g1 = { (int)(2u << 16),          // data_size = 4 bytes
                       (int)(1024u << 16),       // tensor_dim0[15:0] = 1024
                       (int)(1u << 16),          // tensor_dim1[15:0] = 1
                       (int)(1024u << 16),       // tile_dim0 = 1024
                       1,                        // tile_dim1 = 1
                       1024,                     // tensor_dim0_stride[31:0]
                       0, 0 };
    const i32x4 z4 = { 0, 0, 0, 0 };             // groups 2/3 disabled (<=2D)
    const i32x8 z8 = { 0, 0, 0, 0, 0, 0, 0, 0 };

    const unsigned long long gaBase =
        (unsigned long long)(x + (size_t)baseRow * FEAT);

    unsigned m[ROWS_PER_WAVE];

    // Issue phase: one TDM descriptor per unmasked row, back-to-back.
    #pragma unroll
    for (int r = 0; r < ROWS_PER_WAVE; ++r) {
        const unsigned row = baseRow + r;
        m[r] = mask[row & (N_PATCH - 1)];        // uniform scalar load
        if (!m[r]) {
            const unsigned long long ga = gaBase + (unsigned long long)r * (FEAT * 4u);
            const unsigned lds_off =
                (unsigned)(unsigned long long)(&tile[wave][r][0]);  // flat low32 == LDS offset
            u32x4 g0;
            g0.x = rfl(1u);                                  // count = 1 valid descriptor
            g0.y = rfl(lds_off);                             // lds_addr
            g0.z = rfl((unsigned)ga);                        // global_addr[31:0]
            g0.w = rfl((unsigned)((ga >> 32) & 0x01FFFFFFu)  // global_addr[56:32]
                       | 0x80000000u);                       // type = 2 ("image")
            __builtin_amdgcn_tensor_load_to_lds(g0, g1, z4, z4, z8, 0);
        }
    }

    // Masked rows: write zeros now, overlapping the in-flight DMA.
    #pragma unroll
    for (int r = 0; r < ROWS_PER_WAVE; ++r) {
        if (m[r]) {
            f32x4* __restrict__ outv = (f32x4*)(out + (size_t)(baseRow + r) * FEAT);
            const f32x4 z = { 0.f, 0.f, 0.f, 0.f };
            #pragma unroll
            for (int j = 0; j < 8; ++j)
                __builtin_nontemporal_store(z, &outv[j * 32 + lane]);
        }
    }

    // Wait for this wave's TDM transfers, then drain LDS -> global.
    __builtin_amdgcn_s_wait_tensorcnt(0);

    #pragma unroll
    for (int r = 0; r < ROWS_PER_WAVE; ++r) {
        if (!m[r]) {
            f32x4* __restrict__ outv = (f32x4*)(out + (size_t)(baseRow + r) * FEAT);
            const f32x4* lv = (const f32x4*)(&tile[wave][r][0]);
            #pragma unroll
            for (int j = 0; j < 8; ++j)
                __builtin_nontemporal_store(lv[j * 32 + lane], &outv[j * 32 + lane]);
        }
    }
}

// --------------------------------------------------------------------------
// Mask kernel: reproduce jax.random.uniform(key=42, (512,)) via threefry2x32,
// then mask = (stable-sort rank < 204). Trivial cost; one block.
// --------------------------------------------------------------------------
__global__ void PatchMasker_mask_kernel(unsigned* __restrict__ mask) {
    __shared__ float sc[N_PATCH];
    const int t = threadIdx.x;    // 512 threads

    if (t < 256) {
        const unsigned k0 = 0u, k1 = 42u;
        const unsigned k2 = k0 ^ k1 ^ 0x1BD11BDAu;
        unsigned x0 = (unsigned)t + k0;          // counts[0..255]
        unsigned x1 = (unsigned)(t + 256) + k1;  // counts[256..511]
        const int rA[4] = {13, 15, 26, 6};
        const int rB[4] = {17, 29, 16, 24};
        #pragma unroll
        for (int j = 0; j < 4; ++j) { x0 += x1; x1 = rotl32(x1, rA[j]); x1 ^= x0; }
        x0 += k1; x1 += k2 + 1u;
        #pragma unroll
        for (int j = 0; j < 4; ++j) { x0 += x1; x1 = rotl32(x1, rB[j]); x1 ^= x0; }
        x0 += k2; x1 += k0 + 2u;
        #pragma unroll
        for (int j = 0; j < 4; ++j) { x0 += x1; x1 = rotl32(x1, rA[j]); x1 ^= x0; }
        x0 += k0; x1 += k1 + 3u;
        #pragma unroll
        for (int j = 0; j < 4; ++j) { x0 += x1; x1 = rotl32(x1, rB[j]); x1 ^= x0; }
        x0 += k1; x1 += k2 + 4u;
        #pragma unroll
        for (int j = 0; j < 4; ++j) { x0 += x1; x1 = rotl32(x1, rA[j]); x1 ^= x0; }
        x0 += k2; x1 += k0 + 5u;
        // bits -> uniform [0,1): bitcast((bits>>9)|0x3f800000) - 1.0
        sc[t]       = __uint_as_float((x0 >> 9) | 0x3f800000u) - 1.0f;
        sc[t + 256] = __uint_as_float((x1 >> 9) | 0x3f800000u) - 1.0f;
    }
    __syncthreads();

    const float mys = sc[t];
    int rank = 0;
    for (int j = 0; j < N_PATCH; ++j) {
        const float s = sc[j];
        rank += (s < mys || (s == mys && j < t)) ? 1 : 0;   // stable argsort rank
    }
    mask[t] = (rank < N_MASK) ? 1u : 0u;
}

extern "C" void kernel_launch(void* const* d_in, const int* in_sizes, int n_in,
                              void* d_out, int out_size, void* d_ws, size_t ws_size,
                              hipStream_t stream) {
    (void)in_sizes; (void)n_in; (void)out_size; (void)ws_size;
    const float* x   = (const float*)d_in[0];
    unsigned*    msk = (unsigned*)d_ws;          // 512 * 4 bytes of scratch
    float*       out = (float*)d_out;

    PatchMasker_mask_kernel<<<1, N_PATCH, 0, stream>>>(msk);
    PatchMasker_apply_kernel<<<N_ROWS / ROWS_PER_WG, WAVES_PER_WG * 32, 0, stream>>>(
        x, msk, out);
}